// CenterLoss_46651934769258
// MI455X (gfx1250) — compile-verified
//
#include <hip/hip_runtime.h>
#include <hip/hip_bf16.h>

// CenterLoss on MI455X (gfx1250, wave32).
// Pass 1: one wave per 16-row tile. Per K-chunk of 4 features each lane loads
//         8B of inputs + 8B of the gathered center row, forms diff, and feeds
//         the SAME register pair as A and B of V_WMMA_F32_16X16X4_F32.
//         Accumulated over K=256 this produces the 16x16 Gram matrix of the
//         diff tile; its diagonal is the per-row squared distance (full f32).
// Pass 2: deterministic single-block reduction -> mean + clamp constant.

typedef __attribute__((ext_vector_type(2))) float v2f;
typedef __attribute__((ext_vector_type(8))) float v8f;

#define FEATURE_DIM 256
#define BATCH       4096
#define NUM_CLASSES 131072

__global__ __launch_bounds__(128) void center_dist_wmma(
    const float* __restrict__ inputs,
    const int*   __restrict__ targets,
    const float* __restrict__ center,
    float*       __restrict__ dist)
{
    const int lane = threadIdx.x & 31;
    const int wave = threadIdx.x >> 5;
    const int gw   = blockIdx.x * 4 + wave;   // 4 waves per 128-thread block
    const int row0 = gw << 4;                 // 16 batch rows per wave
    const int r    = lane & 15;               // row within the tile (M index)
    const int row  = row0 + r;
    const int koff = (lane >> 4) << 1;        // lanes 0-15 -> K+0/1, lanes 16-31 -> K+2/3

    const int t = targets[row];               // gather index (uniform per M-row pair)
    const float* __restrict__ xp = inputs + (size_t)row * FEATURE_DIM + koff;
    const float* __restrict__ cp = center + (size_t)t   * FEATURE_DIM + koff;

    v8f acc = {};                             // 16x16 f32 Gram accumulator (8 VGPRs)
#pragma unroll 8
    for (int k = 0; k < FEATURE_DIM; k += 4) {
        float2 x = *(const float2*)(xp + k);  // global_load_b64
        float2 c = *(const float2*)(cp + k);  // global_load_b64 (gathered row)
        v2f a;
        a[0] = x.x - c.x;
        a[1] = x.y - c.y;
        // A == B register pair: D += A * A^T  (16x16x4 f32 WMMA)
        acc = __builtin_amdgcn_wmma_f32_16x16x4_f32(
            /*neg_a=*/false, a, /*neg_b=*/false, a,
            /*c_mod=*/(short)0, acc, /*reuse_a=*/false, /*reuse_b=*/false);
    }

    // Extract diagonal D[m,m].
    // D layout: VGPR v, lanes 0-15 -> (M=v, N=lane); lanes 16-31 -> (M=v+8, N=lane-16).
    // D[m,m] sits at (VGPR m, lane m) for m<8 and (VGPR m-8, lane m+16) for m>=8.
    // Every source lane s exposes acc[s & 7]; dest lane m pulls from s = m or m+16.
    const int e = lane & 7;
    float w = acc[0];
    w = (e == 1) ? acc[1] : w;
    w = (e == 2) ? acc[2] : w;
    w = (e == 3) ? acc[3] : w;
    w = (e == 4) ? acc[4] : w;
    w = (e == 5) ? acc[5] : w;
    w = (e == 6) ? acc[6] : w;
    w = (e == 7) ? acc[7] : w;

    const int src = (r < 8) ? r : (r + 16);
    float d2 = __shfl(w, src, 32);

    if (lane < 16) {                          // one distance per tile row
        float d = sqrtf(d2);
        d = fminf(fmaxf(d, 1e-12f), 1e12f);
        dist[row0 + lane] = d;
    }
}

__global__ __launch_bounds__(256) void center_loss_reduce(
    const float* __restrict__ dist,
    float*       __restrict__ out)
{
    __shared__ float s[256];
    const int tid = threadIdx.x;
    float v = 0.0f;
#pragma unroll
    for (int i = tid; i < BATCH; i += 256)    // fixed order -> deterministic
        v += dist[i];
    s[tid] = v;
    __syncthreads();
    for (int off = 128; off > 0; off >>= 1) {
        if (tid < off) s[tid] += s[tid + off];
        __syncthreads();
    }
    if (tid == 0) {
        // mean distance + constant from clamping (C-1) masked zeros to 1e-12
        out[0] = s[0] * (1.0f / (float)BATCH)
               + (float)(NUM_CLASSES - 1) * 1e-12f;
    }
}

extern "C" void kernel_launch(void* const* d_in, const int* in_sizes, int n_in,
                              void* d_out, int out_size, void* d_ws, size_t ws_size,
                              hipStream_t stream) {
    const float* inputs  = (const float*)d_in[0];   // [4096, 256] f32
    const int*   targets = (const int*)  d_in[1];   // [4096] int
    const float* center  = (const float*)d_in[2];   // [131072, 256] f32
    float* ws  = (float*)d_ws;                      // 4096 per-row distances
    float* out = (float*)d_out;                     // scalar f32

    // 4096 rows / 16 rows-per-wave = 256 waves = 64 blocks x 4 waves
    center_dist_wmma<<<BATCH / 64, 128, 0, stream>>>(inputs, targets, center, ws);
    center_loss_reduce<<<1, 256, 0, stream>>>(ws, out);
}